// RelRepWindowContext_43147241455988
// MI455X (gfx1250) — compile-verified
//
#include <hip/hip_runtime.h>
#include <hip/hip_bf16.h>

// Problem constants (match reference)
#define Wn 10
#define NEG_LIMIT -1e9f
#define Bb 2
#define Ss 384
#define Hh 512
#define Kk 24
#define INNER 2304
#define THREEH 1536
#define Mrows (Bb * Kk * Kk)   // 1152

// GEMM tiling
#define BM 64
#define BN 128
#define BK 16
#define ASTRIDE 20    // padded LDS row stride for A (floats); 80B = 16B-aligned
#define BSTRIDE 132   // padded LDS row stride for B (floats); 528B = 16B-aligned

typedef __attribute__((ext_vector_type(2))) float v2f;
typedef __attribute__((ext_vector_type(8))) float v8f;

// ---------------------------------------------------------------------------
// gfx1250 async global->LDS copy (ASYNCcnt-tracked) + wait, via inline asm.
// VDST = per-lane LDS byte address, VADDR = per-lane 64-bit global address.
// ---------------------------------------------------------------------------
__device__ __forceinline__ void async_copy_b128(unsigned lds_byte_addr,
                                                const float* gptr) {
    asm volatile("global_load_async_to_lds_b128 %0, %1, off"
                 :: "v"(lds_byte_addr), "v"(gptr)
                 : "memory");
}
__device__ __forceinline__ void wait_async0() {
    asm volatile("s_wait_asynccnt 0x0" ::: "memory");
}
__device__ __forceinline__ unsigned lds_addr_of(const void* p) {
    // Generic pointer to LDS: low 32 bits carry the LDS byte offset.
    return (unsigned)(unsigned long long)p;
}

// ---------------------------------------------------------------------------
// Kernel 1: per (b,i,j) pair, masked window-max over <=40 candidate tokens,
// fallback to no_context_embedding, write fused [head|tail|ctx] row (3H wide).
// ---------------------------------------------------------------------------
__global__ __launch_bounds__(256) void build_rel_kernel(
    const float* __restrict__ token_reps,          // (B,S,H)
    const unsigned char* __restrict__ token_masks, // (B,S) bool
    const int* __restrict__ span_ids,              // (B,K,2)
    const float* __restrict__ span_reps,           // (B,K,H)
    const float* __restrict__ noctx,               // (H,)
    float* __restrict__ rel)                       // (B*K*K, 3H)
{
    const int p  = blockIdx.x;
    const int b  = p / (Kk * Kk);
    const int ij = p % (Kk * Kk);
    const int i  = ij / Kk;
    const int j  = ij % Kk;

    const int si = span_ids[((b * Kk) + i) * 2 + 0];
    const int ei = span_ids[((b * Kk) + i) * 2 + 1];
    const int sj = span_ids[((b * Kk) + j) * 2 + 0];
    const int ej = span_ids[((b * Kk) + j) * 2 + 1];

    const unsigned char* msk = token_masks + (size_t)b * Ss;
    const float* trb = token_reps + (size_t)b * Ss * Hh;

    const int h0 = threadIdx.x;
    const int h1 = threadIdx.x + 256;

    float m0 = NEG_LIMIT, m1 = NEG_LIMIT;
    bool any = false;

    for (int d = 1; d <= Wn; ++d) {
        int ts[4] = { si - d, ei + d, sj - d, ej + d };
#pragma unroll
        for (int q = 0; q < 4; ++q) {
            int t = ts[q];
            bool ok = (t >= 0) & (t < Ss);
            ok = ok && !(t >= si && t <= ei);
            ok = ok && !(t >= sj && t <= ej);
            ok = ok && (msk[ok ? t : 0] != 0) && ok;
            if (ok) {
                any = true;
                const float* tr = trb + (size_t)t * Hh;
                m0 = fmaxf(m0, tr[h0]);
                m1 = fmaxf(m1, tr[h1]);
            }
        }
    }

    const float c0 = any ? m0 : noctx[h0];
    const float c1 = any ? m1 : noctx[h1];

    float* row = rel + (size_t)p * THREEH;
    const float* hrep = span_reps + (size_t)(b * Kk + i) * Hh;
    const float* trep = span_reps + (size_t)(b * Kk + j) * Hh;

    row[h0]          = hrep[h0];
    row[h1]          = hrep[h1];
    row[Hh + h0]     = trep[h0];
    row[Hh + h1]     = trep[h1];
    row[2 * Hh + h0] = c0;
    row[2 * Hh + h1] = c1;
}

// ---------------------------------------------------------------------------
// f32 WMMA GEMM with double-buffered async-to-LDS staging.
//   C(MxN) = A(MxK) * B(KxN) + bias, optional ReLU.  All row-major f32.
// Block: 256 threads (8 waves), block tile 64x128, K-stage 16.
// Wave grid 2x4; each wave computes a 32x32 output (2x2 WMMA tiles).
//
// WMMA f32 16x16x4 fragment layout (ISA 7.12.2):
//   A: lanes 0-15 -> M=lane, v0/v1 = K0/K1;  lanes 16-31 -> M=lane-16, K2/K3
//   B: lanes 0-15 -> N=lane, v0/v1 = K0/K1;  lanes 16-31 -> N=lane-16, K2/K3
//   C/D: vgpr r -> (M = r + 8*half, N = lane&15)
// ---------------------------------------------------------------------------
template <int RELU>
__global__ __launch_bounds__(256) void wmma_gemm_async(
    const float* __restrict__ A, const float* __restrict__ Bm,
    const float* __restrict__ bias, float* __restrict__ C,
    int M, int N, int Kd)
{
    __shared__ float As[2][BM * ASTRIDE];   // 2 x 5120 B
    __shared__ float Bs[2][BK * BSTRIDE];   // 2 x 8448 B

    const int tid  = threadIdx.x;
    const int wave = tid >> 5;
    const int lane = tid & 31;
    const int half = lane >> 4;   // 0|1
    const int lr   = lane & 15;

    const int m0 = blockIdx.y * BM;
    const int n0 = blockIdx.x * BN;

    const int wm = wave >> 2;     // 0..1  (M direction, 32 rows each)
    const int wn = wave & 3;      // 0..3  (N direction, 32 cols each)

    // Stage-copy thread mapping (one b128 per A slot, two per B):
    const int arow = tid >> 2;          // 0..63
    const int acol = (tid & 3) << 2;    // 0,4,8,12
    const int brow = tid >> 5;          // 0..7 (and +8)
    const int bcol = (tid & 31) << 2;   // 0..124 step 4

    const float* agp  = A  + (size_t)(m0 + arow) * Kd + acol;
    const float* bgp0 = Bm + (size_t)brow       * N + n0 + bcol;
    const float* bgp1 = Bm + (size_t)(brow + 8) * N + n0 + bcol;

    const unsigned aLds0 = lds_addr_of(&As[0][arow * ASTRIDE + acol]);
    const unsigned aLds1 = lds_addr_of(&As[1][arow * ASTRIDE + acol]);
    const unsigned bLds0 = lds_addr_of(&Bs[0][brow * BSTRIDE + bcol]);
    const unsigned bLds1 = lds_addr_of(&Bs[1][brow * BSTRIDE + bcol]);
    const unsigned bStep = 8u * BSTRIDE * sizeof(float);

    auto issue_stage = [&](int buf, int k0) {
        async_copy_b128(buf ? aLds1 : aLds0, agp + k0);
        async_copy_b128(buf ? bLds0 + 0 : bLds0,        // placeholder, fixed below
                        bgp0);
    };
    (void)issue_stage; // (not used; explicit issue below for clarity)

    v8f acc[2][2];
    {
        v8f z = {0.f, 0.f, 0.f, 0.f, 0.f, 0.f, 0.f, 0.f};
        acc[0][0] = z; acc[0][1] = z; acc[1][0] = z; acc[1][1] = z;
    }

    const int nstages = Kd / BK;

    // Prologue: stage 0 into buffer 0.
    async_copy_b128(aLds0, agp);
    async_copy_b128(bLds0, bgp0);
    async_copy_b128(bLds0 + bStep, bgp1);

    for (int s = 0; s < nstages; ++s) {
        wait_async0();        // stage s resident in LDS (this thread's copies)
        __syncthreads();      // ...and everyone else's; prev buffer drained

        const int cur = s & 1;
        if (s + 1 < nstages) {
            const int k0 = (s + 1) * BK;
            const unsigned aL = cur ? aLds0 : aLds1;   // next buffer
            const unsigned bL = cur ? bLds0 : bLds1;
            async_copy_b128(aL, agp + k0);
            async_copy_b128(bL, bgp0 + (size_t)k0 * N);
            async_copy_b128(bL + bStep, bgp1 + (size_t)k0 * N);
        }

        const float* Ab = &As[cur][0];
        const float* Bp = &Bs[cur][0];

#pragma unroll
        for (int kk = 0; kk < BK; kk += 4) {
            v2f a[2], bf[2];
#pragma unroll
            for (int sm = 0; sm < 2; ++sm) {
                const int row = wm * 32 + sm * 16 + lr;
                a[sm].x = Ab[row * ASTRIDE + kk + 2 * half + 0];
                a[sm].y = Ab[row * ASTRIDE + kk + 2 * half + 1];
            }
#pragma unroll
            for (int sn = 0; sn < 2; ++sn) {
                const int col = wn * 32 + sn * 16 + lr;
                bf[sn].x = Bp[(kk + 2 * half + 0) * BSTRIDE + col];
                bf[sn].y = Bp[(kk + 2 * half + 1) * BSTRIDE + col];
            }
#pragma unroll
            for (int sm = 0; sm < 2; ++sm)
#pragma unroll
                for (int sn = 0; sn < 2; ++sn)
                    acc[sm][sn] = __builtin_amdgcn_wmma_f32_16x16x4_f32(
                        false, a[sm], false, bf[sn],
                        (short)0, acc[sm][sn], false, false);
        }
    }

    // Epilogue: bias (+ReLU) and store.
#pragma unroll
    for (int sm = 0; sm < 2; ++sm) {
#pragma unroll
        for (int sn = 0; sn < 2; ++sn) {
            const int col  = n0 + wn * 32 + sn * 16 + lr;
            const int rowb = m0 + wm * 32 + sm * 16 + 8 * half;
            const float bv = bias[col];
            float* crow = C + (size_t)rowb * N + col;
#pragma unroll
            for (int r = 0; r < 8; ++r) {
                float v = acc[sm][sn][r] + bv;
                if (RELU) v = fmaxf(v, 0.0f);
                crow[(size_t)r * N] = v;
            }
        }
    }
}

// ---------------------------------------------------------------------------
extern "C" void kernel_launch(void* const* d_in, const int* in_sizes, int n_in,
                              void* d_out, int out_size, void* d_ws, size_t ws_size,
                              hipStream_t stream) {
    const float*         token_reps  = (const float*)d_in[0];
    const unsigned char* token_masks = (const unsigned char*)d_in[1];
    const int*           span_ids    = (const int*)d_in[2];
    const float*         span_reps   = (const float*)d_in[3];
    const float*         noctx       = (const float*)d_in[4];
    const float*         w1          = (const float*)d_in[5];
    const float*         b1          = (const float*)d_in[6];
    const float*         w2          = (const float*)d_in[7];
    const float*         b2          = (const float*)d_in[8];
    float* out = (float*)d_out;

    float* rel = (float*)d_ws;                   // (1152, 1536)
    float* hid = rel + (size_t)Mrows * THREEH;   // (1152, 2304)

    // 1) fused [head|tail|ctx] rows
    build_rel_kernel<<<Mrows, 256, 0, stream>>>(
        token_reps, token_masks, span_ids, span_reps, noctx, rel);

    // 2) hid = relu(rel @ w1 + b1): (1152x1536)x(1536x2304)
    {
        dim3 grid(INNER / BN, Mrows / BM);       // (18, 18)
        wmma_gemm_async<1><<<grid, 256, 0, stream>>>(
            rel, w1, b1, hid, Mrows, INNER, THREEH);
    }

    // 3) out = hid @ w2 + b2: (1152x2304)x(2304x512)
    {
        dim3 grid(Hh / BN, Mrows / BM);          // (4, 18)
        wmma_gemm_async<0><<<grid, 256, 0, stream>>>(
            hid, w2, b2, out, Mrows, Hh, INNER);
    }
}